// DMI_loss_17600775979806
// MI455X (gfx1250) — compile-verified
//
#include <hip/hip_runtime.h>
#include <math.h>

// Problem constants from the reference: B=64, C=1, H=W=512.
#define NBATCH 64
#define PIX (512 * 512)          // 262144 pixels per batch
#define BPB 16                   // blocks per batch
#define TPB 256                  // threads per block (8 waves of 32 on gfx1250)
#define CHUNK (PIX / BPB)        // 16384 floats per block
#define ITERS (CHUNK / (TPB * 4))// 16 float4 iterations per thread

typedef float v2f __attribute__((ext_vector_type(2)));
typedef float v4f __attribute__((ext_vector_type(4)));
typedef float v8f __attribute__((ext_vector_type(8)));

// Sum 256 f32 values sitting in LDS using the f32 matrix pipe.
// A = ones(16x4); B tiles get the 256 values via a bijective packing
// (exact mapping inside the tile is irrelevant for a total sum).
// D = sum_c ones * B_c  => every row of D holds the 16 column sums;
// D vgpr0 across the 32 lanes = [colsum_0..15, colsum_0..15], so a full
// 32-lane xor-shuffle tree gives 2 * total. Exact f32 arithmetic.
__device__ __forceinline__ float wmma_sum256(const float* sh, int lane) {
  v2f a;
  a.x = 1.0f;
  a.y = 1.0f;
  v8f acc = {};
#pragma unroll
  for (int c = 0; c < 4; ++c) {
    v2f bv;
    bv.x = sh[c * 64 + lane];        // lanes 0..31
    bv.y = sh[c * 64 + 32 + lane];   // lanes 0..31 (+32)
    // 8 args: (neg_a, A, neg_b, B, c_mod, C, reuse_a, reuse_b)
    acc = __builtin_amdgcn_wmma_f32_16x16x4_f32(
        false, a, false, bv, (short)0, acc, false, false);
  }
  float v = acc[0];
#pragma unroll
  for (int m = 16; m >= 1; m >>= 1) v += __shfl_xor(v, m, 32);
  return 0.5f * v;
}

// Stage 1: streaming reduction. Each block owns a contiguous 64KB slice of
// one batch. Non-temporal b128 loads (no reuse; keep the 128MB stream from
// thrashing L2), f32 FMA accumulation, then WMMA-based block reduction.
__global__ void __launch_bounds__(TPB)
dmi_partial_kernel(const float* __restrict__ x, const float* __restrict__ y,
                   float* __restrict__ ws) {
  const int b = blockIdx.y;   // batch
  const int k = blockIdx.x;   // chunk within batch
  const int t = threadIdx.x;
  const size_t base = (size_t)b * PIX + (size_t)k * CHUNK;
  const v4f* __restrict__ xp = reinterpret_cast<const v4f*>(x + base);
  const v4f* __restrict__ yp = reinterpret_cast<const v4f*>(y + base);

  float sxy = 0.0f, sx = 0.0f, sy = 0.0f;
#pragma unroll
  for (int i = 0; i < ITERS; ++i) {
    const int idx = i * TPB + t;   // coalesced: wave reads contiguous 16B*32
    v4f xv = __builtin_nontemporal_load(xp + idx);
    v4f yv = __builtin_nontemporal_load(yp + idx);
    sxy = fmaf(xv.x, yv.x, sxy);
    sxy = fmaf(xv.y, yv.y, sxy);
    sxy = fmaf(xv.z, yv.z, sxy);
    sxy = fmaf(xv.w, yv.w, sxy);
    sx += (xv.x + xv.y) + (xv.z + xv.w);
    sy += (yv.x + yv.y) + (yv.z + yv.w);
  }

  __shared__ float sh[3][TPB];
  sh[0][t] = sxy;
  sh[1][t] = sx;
  sh[2][t] = sy;
  __syncthreads();

  // Every wave redundantly reduces all 256 partials (branch-free => EXEC is
  // all-ones for the WMMAs, as the ISA requires). Lane-indexed LDS reads are
  // bank-conflict free. Only thread 0 publishes.
  const int lane = t & 31;
  const float t_sxy = wmma_sum256(sh[0], lane);
  const float t_sx  = wmma_sum256(sh[1], lane);
  const float t_sy  = wmma_sum256(sh[2], lane);

  if (t == 0) {
    const int blk = b * BPB + k;
    ws[blk * 3 + 0] = t_sxy;
    ws[blk * 3 + 1] = t_sx;
    ws[blk * 3 + 2] = t_sy;
  }
}

// Stage 2: one thread per batch folds its 16 block partials (fixed order =>
// deterministic), builds the 2x2 det, the -log(|det|+eps) loss, and the mean.
__global__ void __launch_bounds__(NBATCH)
dmi_finalize_kernel(const float* __restrict__ ws, float* __restrict__ out) {
  const int b = threadIdx.x;  // 0..63
  float sxy = 0.0f, sx = 0.0f, sy = 0.0f;
#pragma unroll
  for (int k = 0; k < BPB; ++k) {
    const int blk = b * BPB + k;
    sxy += ws[blk * 3 + 0];
    sx  += ws[blk * 3 + 1];
    sy  += ws[blk * 3 + 2];
  }
  const float P = (float)PIX;
  const float m00 = sxy;
  const float m01 = sx - sxy;
  const float m10 = sy - sxy;
  const float m11 = P - sx - sy + sxy;
  const float det = m00 * m11 - m01 * m10;
  const float loss = -logf(fabsf(det) + 0.001f);

  __shared__ float l[NBATCH];
  l[b] = loss;
  __syncthreads();
  if (b == 0) {
    float s = 0.0f;
#pragma unroll
    for (int i = 0; i < NBATCH; ++i) s += l[i];
    out[0] = s * (1.0f / (float)NBATCH);
  }
}

extern "C" void kernel_launch(void* const* d_in, const int* in_sizes, int n_in,
                              void* d_out, int out_size, void* d_ws, size_t ws_size,
                              hipStream_t stream) {
  (void)in_sizes; (void)n_in; (void)out_size; (void)ws_size;
  const float* x = (const float*)d_in[0];   // input  [64,1,512,512] f32
  const float* y = (const float*)d_in[1];   // target [64,1,512,512] f32
  float* ws = (float*)d_ws;                 // 64*16*3 floats = 12 KB
  float* out = (float*)d_out;               // scalar f32

  dim3 grid(BPB, NBATCH);
  dmi_partial_kernel<<<grid, TPB, 0, stream>>>(x, y, ws);
  dmi_finalize_kernel<<<1, NBATCH, 0, stream>>>(ws, out);
}